// DifferentiableExtrusionFast_43301860278301
// MI455X (gfx1250) — compile-verified
//
#include <hip/hip_runtime.h>
#include <math.h>

#define VXS   128
#define NPOLY 32
#define PVERT 16
#define SHARP 100.0f
#define EPSF  1e-8f

typedef __attribute__((ext_vector_type(2))) float v2f;
typedef __attribute__((ext_vector_type(8))) float v8f;

// One block = one batch-slice of 256 pixels (2 image rows).
// Phase 0: threads 0..31 preprocess one polygon each into LDS.
// Phase 1: each thread computes combined mask for its pixel.
// Phase 2: 8 waves extrude via V_WMMA_F32_16X16X4_F32 outer products
//          (depth_mask column  x  16-pixel mask row) and store 33.5MB out.
__global__ __launch_bounds__(256) void extrusion_wmma_kernel(
    const float* __restrict__ polygons,   // (B,32,16,2)
    const float* __restrict__ attributes, // (B,1)
    const float* __restrict__ validity,   // (B,32)
    float* __restrict__ out)              // (B,128,128,128)
{
    __shared__ float s_pv[NPOLY][PVERT][2];   // compacted vertices
    __shared__ float s_e [NPOLY][PVERT][6];   // v0x v0y v1x v1y 1/enorm 1/(dy+eps)
    __shared__ int   s_K [NPOLY];
    __shared__ int   s_val[NPOLY];
    __shared__ float s_comb[256];

    const int b         = blockIdx.x >> 6;        // 64 blocks per batch
    const int pixelBase = (blockIdx.x & 63) * 256;
    const int tid       = threadIdx.x;

    // ---------------- phase 0: polygon preprocessing ----------------
    if (tid < NPOLY) {
        const float* poly = polygons + (size_t)(b * NPOLY + tid) * (PVERT * 2);
        int K = 0;
        for (int i = 0; i < PVERT; ++i) {
            float x = poly[2 * i], y = poly[2 * i + 1];
            if (x + y != 0.0f) {                  // vmask: poly.sum(axis=1) != 0
                s_pv[tid][K][0] = x;
                s_pv[tid][K][1] = y;
                ++K;
            }
        }
        s_K[tid]   = K;
        s_val[tid] = (validity[b * NPOLY + tid] >= 0.5f) ? 1 : 0;
        for (int i = 0; i < K; ++i) {
            int   j   = (i + 1 < K) ? i + 1 : 0;
            float v0x = s_pv[tid][i][0], v0y = s_pv[tid][i][1];
            float v1x = s_pv[tid][j][0], v1y = s_pv[tid][j][1];
            float ex  = v1x - v0x,       ey  = v1y - v0y;
            s_e[tid][i][0] = v0x;
            s_e[tid][i][1] = v0y;
            s_e[tid][i][2] = v1x;
            s_e[tid][i][3] = v1y;
            s_e[tid][i][4] = 1.0f / (ex * ex + ey * ey + EPSF);
            s_e[tid][i][5] = 1.0f / (ey + EPSF);
        }
    }
    __syncthreads();

    // ---------------- phase 1: per-pixel SDF -> sigmoid -> max ----------------
    const int   pix  = pixelBase + tid;
    const float invV = 1.0f / (float)(VXS - 1);
    const float px   = (float)(pix & (VXS - 1)) * invV;   // x = w/127
    const float py   = (float)(pix >> 7) * invV;          // y = h/127

    float comb = 0.0f;
    for (int n = 0; n < NPOLY; ++n) {
        if (!s_val[n]) continue;                 // masked polys contribute 0
        const int K = s_K[n];
        if (K < 3) continue;                     // sdf=1 -> sigmoid(-100) ~ 0
        float md2   = 3.402823e38f;
        int   cross = 0;
        for (int i = 0; i < K; ++i) {
            const float v0x = s_e[n][i][0];
            const float v0y = s_e[n][i][1];
            const float v1x = s_e[n][i][2];
            const float v1y = s_e[n][i][3];
            const float ien = s_e[n][i][4];
            const float idy = s_e[n][i][5];
            const float ex  = v1x - v0x, ey = v1y - v0y;
            const float vx  = px - v0x,  vy = py - v0y;
            float t = (vx * ex + vy * ey) * ien;
            t = fminf(fmaxf(t, 0.0f), 1.0f);
            const float dx = vx - t * ex, dy = vy - t * ey;
            md2 = fminf(md2, dx * dx + dy * dy);
            const bool yc = ((v0y <= py) && (v1y > py)) ||
                            ((v1y <= py) && (v0y > py));
            const float ix = v0x + ex * ((py - v0y) * idy);
            cross += (yc && (ix > px)) ? 1 : 0;
        }
        float sdf = sqrtf(md2);
        if (cross & 1) sdf = -sdf;
        const float m = 1.0f / (1.0f + expf(sdf * SHARP));  // sigmoid(-sdf*100)
        comb = fmaxf(comb, m);
    }
    s_comb[tid] = comb;
    __syncthreads();

    // ---------------- phase 2: WMMA extrusion (rank-1 GEMM) ----------------
    const float norm = fminf(fmaxf(attributes[b], 0.0f), 1.0f);
    int hv = (int)rintf(norm * (float)VXS);      // round-half-even, like jnp.rint
    hv = hv < 1 ? 1 : (hv > VXS ? VXS : hv);

    const int wave = tid >> 5;
    const int lane = tid & 31;
    const int n16  = lane & 15;
    const int half = lane >> 4;
    const bool lo  = (half == 0);

    #pragma unroll
    for (int g = 0; g < 16; ++g) {
        const int tile = wave + (g << 3);        // 0..127 wave-tiles per block
        const int pg   = tile & 15;              // pixel group (16 consecutive w)
        const int dg   = tile >> 4;              // depth group
        const int d0   = dg * 16;

        // A: 16x4, only K=0 column nonzero = depth_mask[d0+m]
        // (16x4 f32 layout: VGPR0 lanes 0-15 hold K=0 for M=0..15)
        v2f a;
        a.x = (lo && ((d0 + n16) < hv)) ? 1.0f : 0.0f;
        a.y = 0.0f;
        // B: 4x16, only row K=0 nonzero = combined[p0+n].
        // Unconditional LDS read (in-bounds for all lanes; upper half
        // re-reads the same words -> broadcast) + cndmask, so EXEC stays
        // all-ones and no divergent ds_load is generated.
        const float cv = s_comb[pg * 16 + n16];
        v2f bb;
        bb.x = lo ? cv : 0.0f;
        bb.y = 0.0f;

        v8f c = {};
        c = __builtin_amdgcn_wmma_f32_16x16x4_f32(false, a, false, bb,
                                                  (short)0, c, false, false);

        // C/D layout: element (M,N) -> VGPR M%8, lane N + 16*(M/8)
        float* o = out + (size_t)b * (size_t)(VXS * VXS * VXS)
                       + (size_t)(d0 + half * 8) * (size_t)(VXS * VXS)
                       + (size_t)(pixelBase + pg * 16 + n16);
        #pragma unroll
        for (int r = 0; r < 8; ++r)
            o[(size_t)r * (VXS * VXS)] = c[r];
    }
}

extern "C" void kernel_launch(void* const* d_in, const int* in_sizes, int n_in,
                              void* d_out, int out_size, void* d_ws, size_t ws_size,
                              hipStream_t stream) {
    (void)n_in; (void)out_size; (void)d_ws; (void)ws_size;
    const float* polygons   = (const float*)d_in[0];
    const float* attributes = (const float*)d_in[1];
    const float* validity   = (const float*)d_in[2];
    float*       out        = (float*)d_out;

    const int B = in_sizes[1];            // attributes is (B,1)
    dim3 grid(B * 64), block(256);
    hipLaunchKernelGGL(extrusion_wmma_kernel, grid, block, 0, stream,
                       polygons, attributes, validity, out);
}